// MultiFraudNet_47528108097884
// MI455X (gfx1250) — compile-verified
//
#include <hip/hip_runtime.h>
#include <hip/hip_bf16.h>
#include <math.h>

// ---------------------------------------------------------------------------
// MultiFraudNet on MI455X (gfx1250).
// All dense math runs on the WMMA fp32 matrix pipe (V_WMMA_F32_16X16X4_F32),
// since every layer in this net has an output width of exactly 16.
// Graph conv uses f32 global atomics; LSTM recurrence is VALU with LDS-cached
// recurrent weights.
// ---------------------------------------------------------------------------

#define NCOMP  50000
#define NTRANS 500000
#define ECOMP  400000
#define ETRANS 2000000
#define HIST   20
#define CFEAT  64
#define TFEAT  32
#define HDIM   16
#define HLH    8     // half hidden (per LSTM direction)

typedef float v2f __attribute__((ext_vector_type(2)));
typedef float v8f __attribute__((ext_vector_type(8)));

__device__ __forceinline__ v8f wmma4(v2f a, v2f b, v8f c) {
  // D = A(16x4 f32) * B(4x16 f32) + C(16x16 f32)
  return __builtin_amdgcn_wmma_f32_16x16x4_f32(false, a, false, b, (short)0, c,
                                               false, false);
}

__device__ __forceinline__ float gelu_exact(float x) {
  return 0.5f * x * (1.0f + erff(x * 0.7071067811865475f));
}
__device__ __forceinline__ float sigm(float x) {
  return 1.0f / (1.0f + expf(-x));
}

// ---------------------------------------------------------------------------
// Generic GEMM: Y[16-wide] = act(X[n,K] @ W[K,16] + b).  One 16x16 tile per
// wave; wave-uniform guard keeps EXEC all-1s for WMMA.
// A layout (16x4 f32): lanes 0-15 carry K=0,1 ; lanes 16-31 carry K=2,3.
// ---------------------------------------------------------------------------
template <int K, bool ACT_GELU>
__global__ void gemm16_kernel(const float* __restrict__ X,
                              const float* __restrict__ W,
                              const float* __restrict__ bias,
                              float* __restrict__ Y, int n_tiles) {
  const int lane = threadIdx.x & 31;
  const int wave = threadIdx.x >> 5;
  const int tile = blockIdx.x * (blockDim.x >> 5) + wave;
  if (tile >= n_tiles) return;  // uniform per wave
  const int row0 = tile << 4;
  const int m    = lane & 15;          // M for A, N for B/C/D
  const int kof  = (lane >> 4) << 1;   // 0 or 2
  const float* xr = X + (size_t)(row0 + m) * K;
  v8f acc = {0.f, 0.f, 0.f, 0.f, 0.f, 0.f, 0.f, 0.f};
#pragma unroll
  for (int kk = 0; kk < K; kk += 4) {
    v2f a, b;
    a.x = xr[kk + kof];
    a.y = xr[kk + kof + 1];
    b.x = W[(kk + kof) * 16 + m];
    b.y = W[(kk + kof + 1) * 16 + m];
    acc = wmma4(a, b, acc);
  }
  const int mb = (lane >> 4) << 3;  // row base 0 or 8 in C/D layout
#pragma unroll
  for (int r = 0; r < 8; ++r) {
    float v = acc[r] + bias[m];
    if (ACT_GELU) v = gelu_exact(v);
    Y[(size_t)(row0 + mb + r) * 16 + m] = v;
  }
}

// ---------------------------------------------------------------------------
// LSTM pre-gates, fused gather + GEMM:
//   pre_{f,b}[r, 0:32] = x_t[idx[r]] * mask @ Wih_{f,b} + bih + bhh
// 4 accumulators (fwd/bwd x 2 column halves) share one A fragment.
// ---------------------------------------------------------------------------
__global__ void lstm_pre_kernel(const float* __restrict__ xt,
                                const int* __restrict__ idx,
                                const float* __restrict__ Wihf,
                                const float* __restrict__ Wihb,
                                const float* __restrict__ bihf,
                                const float* __restrict__ bhhf,
                                const float* __restrict__ bihb,
                                const float* __restrict__ bhhb,
                                float* __restrict__ pref,
                                float* __restrict__ preb, int n_tiles) {
  const int lane = threadIdx.x & 31;
  const int wave = threadIdx.x >> 5;
  const int tile = blockIdx.x * (blockDim.x >> 5) + wave;
  if (tile >= n_tiles) return;
  const int row0 = tile << 4;
  const int m    = lane & 15;
  const int kof  = (lane >> 4) << 1;
  const int r    = row0 + m;
  const int ti   = idx[r];
  const float msk = (ti >= 0) ? 1.0f : 0.0f;
  const float* xr = xt + (size_t)((ti >= 0) ? ti : 0) * HDIM;
  v8f af0 = {0.f,0.f,0.f,0.f,0.f,0.f,0.f,0.f};
  v8f af1 = af0, ab0 = af0, ab1 = af0;
#pragma unroll
  for (int kk = 0; kk < HDIM; kk += 4) {
    v2f a;
    a.x = xr[kk + kof] * msk;
    a.y = xr[kk + kof + 1] * msk;
    v2f bf0, bf1, bb0, bb1;
    bf0.x = Wihf[(kk + kof) * 32 + m];      bf0.y = Wihf[(kk + kof + 1) * 32 + m];
    bf1.x = Wihf[(kk + kof) * 32 + 16 + m]; bf1.y = Wihf[(kk + kof + 1) * 32 + 16 + m];
    bb0.x = Wihb[(kk + kof) * 32 + m];      bb0.y = Wihb[(kk + kof + 1) * 32 + m];
    bb1.x = Wihb[(kk + kof) * 32 + 16 + m]; bb1.y = Wihb[(kk + kof + 1) * 32 + 16 + m];
    af0 = wmma4(a, bf0, af0);
    af1 = wmma4(a, bf1, af1);
    ab0 = wmma4(a, bb0, ab0);
    ab1 = wmma4(a, bb1, ab1);
  }
  const int mb = (lane >> 4) << 3;
#pragma unroll
  for (int r2 = 0; r2 < 8; ++r2) {
    const size_t rr = (size_t)(row0 + mb + r2) * 32;
    pref[rr + m]      = af0[r2] + bihf[m]      + bhhf[m];
    pref[rr + 16 + m] = af1[r2] + bihf[16 + m] + bhhf[16 + m];
    preb[rr + m]      = ab0[r2] + bihb[m]      + bhhb[m];
    preb[rr + 16 + m] = ab1[r2] + bihb[16 + m] + bhhb[16 + m];
  }
}

// ---------------------------------------------------------------------------
// Bidirectional LSTM recurrence: one thread per company; Whh in LDS.
// ---------------------------------------------------------------------------
__global__ void lstm_kernel(const float* __restrict__ pref,
                            const float* __restrict__ preb,
                            const float* __restrict__ Whhf,
                            const float* __restrict__ Whhb,
                            float* __restrict__ hcat, int nc) {
  __shared__ float wf[HLH * 32];
  __shared__ float wb[HLH * 32];
  for (int i = threadIdx.x; i < HLH * 32; i += blockDim.x) {
    wf[i] = Whhf[i];
    wb[i] = Whhb[i];
  }
  __syncthreads();
  const int b = blockIdx.x * blockDim.x + threadIdx.x;
  if (b >= nc) return;

  float h[HLH], c[HLH], g[32];
  // ---- forward ----
#pragma unroll
  for (int k = 0; k < HLH; ++k) { h[k] = 0.f; c[k] = 0.f; }
  for (int t = 0; t < HIST; ++t) {
    const float* p = pref + (size_t)(b * HIST + t) * 32;
#pragma unroll
    for (int j = 0; j < 32; ++j) {
      float s = p[j];
#pragma unroll
      for (int k = 0; k < HLH; ++k) s += h[k] * wf[k * 32 + j];
      g[j] = s;
    }
#pragma unroll
    for (int k = 0; k < HLH; ++k) {
      const float ig = sigm(g[k]);
      const float fg = sigm(g[8 + k]);
      const float gg = tanhf(g[16 + k]);
      const float og = sigm(g[24 + k]);
      c[k] = fg * c[k] + ig * gg;
      h[k] = og * tanhf(c[k]);
      hcat[(size_t)(b * HIST + t) * HDIM + k] = h[k];
    }
  }
  // ---- backward ----
#pragma unroll
  for (int k = 0; k < HLH; ++k) { h[k] = 0.f; c[k] = 0.f; }
  for (int t = HIST - 1; t >= 0; --t) {
    const float* p = preb + (size_t)(b * HIST + t) * 32;
#pragma unroll
    for (int j = 0; j < 32; ++j) {
      float s = p[j];
#pragma unroll
      for (int k = 0; k < HLH; ++k) s += h[k] * wb[k * 32 + j];
      g[j] = s;
    }
#pragma unroll
    for (int k = 0; k < HLH; ++k) {
      const float ig = sigm(g[k]);
      const float fg = sigm(g[8 + k]);
      const float gg = tanhf(g[16 + k]);
      const float og = sigm(g[24 + k]);
      c[k] = fg * c[k] + ig * gg;
      h[k] = og * tanhf(c[k]);
      hcat[(size_t)(b * HIST + t) * HDIM + 8 + k] = h[k];
    }
  }
}

// ---------------------------------------------------------------------------
// Attention pooling + company head (fused): one thread per company.
// ---------------------------------------------------------------------------
__global__ void attn_head_kernel(const float* __restrict__ hcat,
                                 const float* __restrict__ xc,
                                 const float* __restrict__ Wa,
                                 const float* __restrict__ ba,
                                 const float* __restrict__ Wfc,
                                 const float* __restrict__ bfc,
                                 float* __restrict__ out, int nc) {
  const int b = blockIdx.x * blockDim.x + threadIdx.x;
  if (b >= nc) return;
  float wa[HDIM];
#pragma unroll
  for (int k = 0; k < HDIM; ++k) wa[k] = Wa[k];
  float lg[HIST];
  float mx = -3.4e38f;
  for (int t = 0; t < HIST; ++t) {
    const float* hh = hcat + (size_t)(b * HIST + t) * HDIM;
    float s = ba[0];
#pragma unroll
    for (int k = 0; k < HDIM; ++k) s += hh[k] * wa[k];
    lg[t] = s;
    mx = fmaxf(mx, s);
  }
  float se = 0.f;
  for (int t = 0; t < HIST; ++t) {
    lg[t] = expf(lg[t] - mx);
    se += lg[t];
  }
  const float inv = 1.0f / se;
  float z[HDIM];
#pragma unroll
  for (int k = 0; k < HDIM; ++k) z[k] = 0.f;
  for (int t = 0; t < HIST; ++t) {
    const float a = lg[t] * inv;
    const float* hh = hcat + (size_t)(b * HIST + t) * HDIM;
#pragma unroll
    for (int k = 0; k < HDIM; ++k) z[k] += a * hh[k];
  }
  float xs[HDIM];
#pragma unroll
  for (int k = 0; k < HDIM; ++k) xs[k] = xc[(size_t)b * HDIM + k] + z[k];
#pragma unroll
  for (int j = 0; j < 2; ++j) {
    float s = bfc[j];
#pragma unroll
    for (int k = 0; k < HDIM; ++k) s += xs[k] * Wfc[k * 2 + j];
    out[(size_t)b * 2 + j] = s;
  }
}

// ---------------------------------------------------------------------------
// Transaction head: one thread per transaction.
// ---------------------------------------------------------------------------
__global__ void trans_head_kernel(const float* __restrict__ xt,
                                  const float* __restrict__ Wft,
                                  const float* __restrict__ bft,
                                  float* __restrict__ out, int nt) {
  const int i = blockIdx.x * blockDim.x + threadIdx.x;
  if (i >= nt) return;
  float x[HDIM];
  const float4* xr = (const float4*)(xt + (size_t)i * HDIM);
#pragma unroll
  for (int q = 0; q < 4; ++q) {
    float4 v = xr[q];
    x[q * 4 + 0] = v.x; x[q * 4 + 1] = v.y; x[q * 4 + 2] = v.z; x[q * 4 + 3] = v.w;
  }
#pragma unroll
  for (int j = 0; j < 2; ++j) {
    float s = bft[j];
#pragma unroll
    for (int k = 0; k < HDIM; ++k) s += x[k] * Wft[k * 2 + j];
    out[(size_t)i * 2 + j] = s;
  }
}

// ---------------------------------------------------------------------------
// Graph-conv helpers.
// ---------------------------------------------------------------------------
__global__ void fill_kernel(float* __restrict__ p, float v, int n) {
  const int i = blockIdx.x * blockDim.x + threadIdx.x;
  if (i < n) p[i] = v;
}
__global__ void deg_acc_kernel(float* __restrict__ deg,
                               const int* __restrict__ col, int e) {
  const int i = blockIdx.x * blockDim.x + threadIdx.x;
  if (i < e) atomicAdd(&deg[col[i]], 1.0f);
}
__global__ void rsqrt_kernel(float* __restrict__ deg, int n) {
  const int i = blockIdx.x * blockDim.x + threadIdx.x;
  if (i < n) deg[i] = rsqrtf(deg[i]);  // deg >= 1 due to self loop
}
// out[row[e]] += dis[row]*dis[col] * y[col[e]]
__global__ void scatter_kernel(const float* __restrict__ y,
                               const int* __restrict__ row,
                               const int* __restrict__ col,
                               const float* __restrict__ dis,
                               float* __restrict__ out, int e) {
  const int i = blockIdx.x * blockDim.x + threadIdx.x;
  if (i >= e) return;
  const int r = row[i], c = col[i];
  const float nrm = dis[r] * dis[c];
  const float4* ys = (const float4*)(y + (size_t)c * HDIM);
  float* op = out + (size_t)r * HDIM;
#pragma unroll
  for (int q = 0; q < 4; ++q) {
    float4 v = ys[q];
    atomicAdd(&op[q * 4 + 0], nrm * v.x);
    atomicAdd(&op[q * 4 + 1], nrm * v.y);
    atomicAdd(&op[q * 4 + 2], nrm * v.z);
    atomicAdd(&op[q * 4 + 3], nrm * v.w);
  }
}
// self loops, kernel-ordered after scatter => plain RMW
__global__ void selfloop_kernel(const float* __restrict__ y,
                                const float* __restrict__ dis,
                                float* __restrict__ out, int n16) {
  const int i = blockIdx.x * blockDim.x + threadIdx.x;
  if (i >= n16) return;
  const int node = i >> 4;
  const float d = dis[node];
  out[i] += d * d * y[i];
}

// x_c = (x_c + gelu(conv)) / 2
__global__ void combine_company_kernel(float* __restrict__ xc,
                                       const float* __restrict__ conv, int n) {
  const int i = blockIdx.x * blockDim.x + threadIdx.x;
  if (i < n) xc[i] = 0.5f * (xc[i] + gelu_exact(conv[i]));
}
// x1 = gelu(conv); xt = x0 + x1
__global__ void combine1_kernel(const float* __restrict__ conv,
                                const float* __restrict__ x0,
                                float* __restrict__ x1,
                                float* __restrict__ xt, int n) {
  const int i = blockIdx.x * blockDim.x + threadIdx.x;
  if (i < n) {
    const float g = gelu_exact(conv[i]);
    x1[i] = g;
    xt[i] = x0[i] + g;
  }
}
// avg = (xt + x1 + 2*gelu(conv)) / 3   ( == ((xt+x2) + x1 + x2)/3 )
__global__ void combine2_kernel(const float* __restrict__ conv,
                                const float* __restrict__ xt,
                                const float* __restrict__ x1,
                                float* __restrict__ avg, int n) {
  const int i = blockIdx.x * blockDim.x + threadIdx.x;
  if (i < n) {
    const float g = gelu_exact(conv[i]);
    avg[i] = (xt[i] + x1[i] + 2.0f * g) * (1.0f / 3.0f);
  }
}

// ---------------------------------------------------------------------------
// Host-side orchestration.
// ---------------------------------------------------------------------------
static inline int cdiv_i(long a, long b) { return (int)((a + b - 1) / b); }

extern "C" void kernel_launch(void* const* d_in, const int* in_sizes, int n_in,
                              void* d_out, int out_size, void* d_ws,
                              size_t ws_size, hipStream_t stream) {
  // ---- inputs ----
  const float* company_x = (const float*)d_in[0];   // [50000,64]
  const float* trans_x   = (const float*)d_in[1];   // [500000,32]
  const int*   cedge     = (const int*)d_in[2];     // [2,400000]
  const int*   tedge     = (const int*)d_in[3];     // [2,2000000]
  const int*   cidx      = (const int*)d_in[4];     // [50000,20]
  // params flattened in JAX pytree (sorted dict key) order:
  const float* attn_W  = (const float*)d_in[5];   // [16,1]
  const float* attn_b  = (const float*)d_in[6];   // [1]
  const float* cffn_W  = (const float*)d_in[7];   // [64,16]
  const float* cffn_b  = (const float*)d_in[8];   // [16]
  const float* cpost_W = (const float*)d_in[9];   // [16,16]
  const float* cpost_b = (const float*)d_in[10];  // [16]
  const float* convc_W = (const float*)d_in[11];  // [16,16]
  const float* convc_b = (const float*)d_in[12];  // [16]
  const float* fcc_W   = (const float*)d_in[13];  // [16,2]
  const float* fcc_b   = (const float*)d_in[14];  // [2]
  const float* fct_W   = (const float*)d_in[15];  // [16,2]
  const float* fct_b   = (const float*)d_in[16];  // [2]
  const float* lb_Wih  = (const float*)d_in[17];  // [16,32]
  const float* lb_Whh  = (const float*)d_in[18];  // [8,32]
  const float* lb_bih  = (const float*)d_in[19];  // [32]
  const float* lb_bhh  = (const float*)d_in[20];  // [32]
  const float* lf_Wih  = (const float*)d_in[21];
  const float* lf_Whh  = (const float*)d_in[22];
  const float* lf_bih  = (const float*)d_in[23];
  const float* lf_bhh  = (const float*)d_in[24];
  const float* tc1_W   = (const float*)d_in[25];  // [16,16]
  const float* tc1_b   = (const float*)d_in[26];
  const float* tc2_W   = (const float*)d_in[27];
  const float* tc2_b   = (const float*)d_in[28];
  const float* tffn_W  = (const float*)d_in[29];  // [32,16]
  const float* tffn_b  = (const float*)d_in[30];
  const float* tpost_W = (const float*)d_in[31];  // [16,16]
  const float* tpost_b = (const float*)d_in[32];
  (void)in_sizes; (void)n_in; (void)out_size; (void)ws_size;

  // ---- workspace carve (~495 MB, all 16-float aligned) ----
  float* w = (float*)d_ws;
  float* xc    = w; w += (size_t)NCOMP * HDIM;
  float* xc2   = w; w += (size_t)NCOMP * HDIM;
  float* yc    = w; w += (size_t)NCOMP * HDIM;
  float* oc    = w; w += (size_t)NCOMP * HDIM;
  float* degc  = w; w += (size_t)NCOMP;
  float* xt0   = w; w += (size_t)NTRANS * HDIM;
  float* x1b   = w; w += (size_t)NTRANS * HDIM;
  float* xtb   = w; w += (size_t)NTRANS * HDIM;
  float* ytmp  = w; w += (size_t)NTRANS * HDIM;
  float* convo = w; w += (size_t)NTRANS * HDIM;
  float* degt  = w; w += (size_t)NTRANS;
  float* pref  = w; w += (size_t)NCOMP * HIST * 32;
  float* preb  = w; w += (size_t)NCOMP * HIST * 32;
  float* hcat  = w; w += (size_t)NCOMP * HIST * HDIM;

  const int B = 256;                 // 8 waves / block
  const int ct = NCOMP / 16;         // 3125 row tiles (exact)
  const int tt = NTRANS / 16;        // 31250 row tiles (exact)
  const int pt = (NCOMP * HIST) / 16;// 62500 row tiles (exact)

  // ================= company branch =================
  gemm16_kernel<CFEAT, true><<<cdiv_i(ct, 8), B, 0, stream>>>(
      company_x, cffn_W, cffn_b, xc, ct);
  fill_kernel<<<cdiv_i(NCOMP, B), B, 0, stream>>>(degc, 1.0f, NCOMP);
  deg_acc_kernel<<<cdiv_i(ECOMP, B), B, 0, stream>>>(degc, cedge + ECOMP, ECOMP);
  rsqrt_kernel<<<cdiv_i(NCOMP, B), B, 0, stream>>>(degc, NCOMP);
  gemm16_kernel<HDIM, false><<<cdiv_i(ct, 8), B, 0, stream>>>(
      xc, convc_W, convc_b, yc, ct);
  fill_kernel<<<cdiv_i(NCOMP * HDIM, B), B, 0, stream>>>(oc, 0.0f, NCOMP * HDIM);
  scatter_kernel<<<cdiv_i(ECOMP, B), B, 0, stream>>>(
      yc, cedge, cedge + ECOMP, degc, oc, ECOMP);
  selfloop_kernel<<<cdiv_i(NCOMP * HDIM, B), B, 0, stream>>>(yc, degc, oc,
                                                             NCOMP * HDIM);
  combine_company_kernel<<<cdiv_i(NCOMP * HDIM, B), B, 0, stream>>>(
      xc, oc, NCOMP * HDIM);
  gemm16_kernel<HDIM, true><<<cdiv_i(ct, 8), B, 0, stream>>>(
      xc, cpost_W, cpost_b, xc2, ct);

  // ================= transaction branch =================
  gemm16_kernel<TFEAT, true><<<cdiv_i(tt, 8), B, 0, stream>>>(
      trans_x, tffn_W, tffn_b, xt0, tt);
  fill_kernel<<<cdiv_i(NTRANS, B), B, 0, stream>>>(degt, 1.0f, NTRANS);
  deg_acc_kernel<<<cdiv_i(ETRANS, B), B, 0, stream>>>(degt, tedge + ETRANS,
                                                      ETRANS);
  rsqrt_kernel<<<cdiv_i(NTRANS, B), B, 0, stream>>>(degt, NTRANS);
  // conv1
  gemm16_kernel<HDIM, false><<<cdiv_i(tt, 8), B, 0, stream>>>(
      xt0, tc1_W, tc1_b, ytmp, tt);
  fill_kernel<<<cdiv_i(NTRANS * HDIM, B), B, 0, stream>>>(convo, 0.0f,
                                                          NTRANS * HDIM);
  scatter_kernel<<<cdiv_i(ETRANS, B), B, 0, stream>>>(
      ytmp, tedge, tedge + ETRANS, degt, convo, ETRANS);
  selfloop_kernel<<<cdiv_i(NTRANS * HDIM, B), B, 0, stream>>>(ytmp, degt, convo,
                                                              NTRANS * HDIM);
  combine1_kernel<<<cdiv_i(NTRANS * HDIM, B), B, 0, stream>>>(
      convo, xt0, x1b, xtb, NTRANS * HDIM);
  // conv2
  gemm16_kernel<HDIM, false><<<cdiv_i(tt, 8), B, 0, stream>>>(
      xtb, tc2_W, tc2_b, ytmp, tt);
  fill_kernel<<<cdiv_i(NTRANS * HDIM, B), B, 0, stream>>>(convo, 0.0f,
                                                          NTRANS * HDIM);
  scatter_kernel<<<cdiv_i(ETRANS, B), B, 0, stream>>>(
      ytmp, tedge, tedge + ETRANS, degt, convo, ETRANS);
  selfloop_kernel<<<cdiv_i(NTRANS * HDIM, B), B, 0, stream>>>(ytmp, degt, convo,
                                                              NTRANS * HDIM);
  combine2_kernel<<<cdiv_i(NTRANS * HDIM, B), B, 0, stream>>>(
      convo, xtb, x1b, ytmp, NTRANS * HDIM);  // x_avg -> ytmp
  gemm16_kernel<HDIM, true><<<cdiv_i(tt, 8), B, 0, stream>>>(
      ytmp, tpost_W, tpost_b, xt0, tt);  // final x_t -> xt0

  // ================= LSTM + heads =================
  lstm_pre_kernel<<<cdiv_i(pt, 8), B, 0, stream>>>(
      xt0, cidx, lf_Wih, lb_Wih, lf_bih, lf_bhh, lb_bih, lb_bhh, pref, preb, pt);
  lstm_kernel<<<cdiv_i(NCOMP, B), B, 0, stream>>>(pref, preb, lf_Whh, lb_Whh,
                                                  hcat, NCOMP);
  attn_head_kernel<<<cdiv_i(NCOMP, B), B, 0, stream>>>(
      hcat, xc2, attn_W, attn_b, fcc_W, fcc_b, (float*)d_out, NCOMP);
  trans_head_kernel<<<cdiv_i(NTRANS, B), B, 0, stream>>>(
      xt0, fct_W, fct_b, (float*)d_out + 2 * (size_t)NCOMP, NTRANS);
}